// KMeansClustering_2671469658718
// MI455X (gfx1250) — compile-verified
//
#include <hip/hip_runtime.h>

typedef __attribute__((ext_vector_type(2))) float v2f;
typedef __attribute__((ext_vector_type(8))) float v8f;

#define NPTS   131072      // 32 * 4096
#define DIM    128
#define NCLUST 100
#define KPAD   112         // 7 tiles of 16
#define NT     7
#define LSTR   130         // LDS row stride (floats): conflict-free, 8B-aligned rows

// ---------------------------------------------------------------------------
// centers[0:100] = points[0:100]; pad rows zeroed.  grid: 112 x 128
// ---------------------------------------------------------------------------
__global__ void km_init_centers(const float* __restrict__ pts, float* __restrict__ centers) {
    int i = blockIdx.x * DIM + threadIdx.x;            // 0 .. 112*128-1
    centers[i] = (i < NCLUST * DIM) ? pts[i] : 0.0f;
}

// ---------------------------------------------------------------------------
// c2[k] = ||centers[k]||^2 (pad -> +huge), zero sums/counts.  grid: 112 x 128
// ---------------------------------------------------------------------------
__global__ void km_c2_zero(const float* __restrict__ centers, float* __restrict__ c2,
                           float* __restrict__ sums, float* __restrict__ counts) {
    int k = blockIdx.x, d = threadIdx.x;
    float v = centers[k * DIM + d];
    if (k < NCLUST) sums[k * DIM + d] = 0.0f;
    float sq = v * v;
    #pragma unroll
    for (int m = 1; m < 32; m <<= 1) sq += __shfl_xor(sq, m, 32);
    __shared__ float part[4];
    if ((d & 31) == 0) part[d >> 5] = sq;
    __syncthreads();
    if (d == 0) {
        float s = part[0] + part[1] + part[2] + part[3];
        c2[k]     = (k < NCLUST) ? s : 3.0e38f;
        counts[k] = 0.0f;
    }
}

// ---------------------------------------------------------------------------
// Assignment via V_WMMA_F32_16X16X4_F32.
// Block: 128 threads = 4 waves; each wave computes a 16-point x 112-cluster
// dot-product tile (K=128 in chunks of 4), centers resident in LDS.
// argmin_k ( c2[k] - 2*dot )  ==  argmin_k ||p - c_k||^2  (p^2 const).
// grid: NPTS/64 blocks.
// ---------------------------------------------------------------------------
__global__ void __launch_bounds__(128)
km_assign(const float* __restrict__ pts, const float* __restrict__ centers,
          const float* __restrict__ c2, int* __restrict__ assign) {
    __shared__ float sC[KPAD * LSTR];
    __shared__ float sC2[KPAD];

    const int tid = threadIdx.x;
    // cooperative load of padded centers into LDS (row stride 130)
    for (int idx = tid; idx < KPAD * DIM; idx += 128) {
        int r = idx >> 7, c = idx & (DIM - 1);
        sC[r * LSTR + c] = centers[idx];
    }
    if (tid < KPAD) sC2[tid] = c2[tid];
    __syncthreads();

    const int wave = tid >> 5;
    const int lane = tid & 31;
    const int hi   = lane >> 4;        // 0 | 1  -> K sub-pair, M/N +8 half
    const int lo   = lane & 15;        // M (for A) / N (for B) within tile
    const int base = blockIdx.x * 64 + wave * 16;

    v8f acc[NT];
    #pragma unroll
    for (int t = 0; t < NT; ++t) acc[t] = (v8f)(0.0f);

    const float* prow = pts + (size_t)(base + lo) * DIM;
    __builtin_prefetch(prow, 0, 3);    // global_prefetch_b8

    #pragma unroll 4
    for (int k = 0; k < DIM; k += 4) {
        const int kk = k + 2 * hi;
        // A fragment: 16x4 f32 — lane lo = row M, v0/v1 = K = 2*hi, 2*hi+1
        v2f a;
        a.x = prow[kk];
        a.y = prow[kk + 1];
        #pragma unroll
        for (int t = 0; t < NT; ++t) {
            // B fragment: 4x16 f32 — lane lo = col N, K mirrored with A
            const float* crow = &sC[(t * 16 + lo) * LSTR + kk];
            v2f b;
            b.x = crow[0];
            b.y = crow[1];
            acc[t] = __builtin_amdgcn_wmma_f32_16x16x4_f32(
                false, a, false, b, (short)0, acc[t], false, false);
        }
    }

    // Epilogue: D layout — VGPR r: lanes 0-15 -> M=r, lanes 16-31 -> M=r+8.
    #pragma unroll
    for (int r = 0; r < 8; ++r) {
        float bestv = 3.4e38f;
        int   besti = 0;
        #pragma unroll
        for (int t = 0; t < NT; ++t) {
            int   n = t * 16 + lo;
            float v = sC2[n] - 2.0f * acc[t][r];
            if (v < bestv) { bestv = v; besti = n; }
        }
        // reduce argmin across the 16 lanes of each half-wave
        #pragma unroll
        for (int m = 1; m < 16; m <<= 1) {
            float ov = __shfl_xor(bestv, m, 32);
            int   oi = __shfl_xor(besti, m, 32);
            if (ov < bestv || (ov == bestv && oi < besti)) { bestv = ov; besti = oi; }
        }
        if (lo == 0) assign[base + r + 8 * hi] = besti;
    }
}

// ---------------------------------------------------------------------------
// Accumulate sums/counts: LDS f32 atomics per block, then global f32 atomics.
// Block: 256 threads (8 waves), 512 points/block.  grid: NPTS/512.
// ---------------------------------------------------------------------------
__global__ void __launch_bounds__(256)
km_accum(const float* __restrict__ pts, const int* __restrict__ assign,
         float* __restrict__ sums, float* __restrict__ counts) {
    __shared__ float ls[NCLUST * DIM];
    __shared__ float lc[NCLUST];

    const int tid = threadIdx.x;
    for (int i = tid; i < NCLUST * DIM; i += 256) ls[i] = 0.0f;
    for (int i = tid; i < NCLUST; i += 256) lc[i] = 0.0f;
    __syncthreads();

    const int wave = tid >> 5, lane = tid & 31;
    const int base = blockIdx.x * 512;
    for (int p = wave; p < 512; p += 8) {
        const int gi = base + p;
        const int a  = assign[gi];
        const float* row = pts + (size_t)gi * DIM;
        float* dst = &ls[a * DIM];
        #pragma unroll
        for (int j = 0; j < 4; ++j) {
            int d = lane + j * 32;
            __hip_atomic_fetch_add(&dst[d], row[d],
                                   __ATOMIC_RELAXED, __HIP_MEMORY_SCOPE_WORKGROUP);
        }
        if (lane == 0)
            __hip_atomic_fetch_add(&lc[a], 1.0f,
                                   __ATOMIC_RELAXED, __HIP_MEMORY_SCOPE_WORKGROUP);
    }
    __syncthreads();

    for (int i = tid; i < NCLUST * DIM; i += 256)
        __hip_atomic_fetch_add(&sums[i], ls[i],
                               __ATOMIC_RELAXED, __HIP_MEMORY_SCOPE_AGENT);
    for (int i = tid; i < NCLUST; i += 256)
        __hip_atomic_fetch_add(&counts[i], lc[i],
                               __ATOMIC_RELAXED, __HIP_MEMORY_SCOPE_AGENT);
}

// ---------------------------------------------------------------------------
// centers[k] = counts[k] > 0 ? sums[k]/max(counts[k],1) : centers[k]
// grid: 100 x 128
// ---------------------------------------------------------------------------
__global__ void km_update(const float* __restrict__ sums, const float* __restrict__ counts,
                          float* __restrict__ centers) {
    int k = blockIdx.x, d = threadIdx.x;
    float c = counts[k];
    float old = centers[k * DIM + d];
    centers[k * DIM + d] = (c > 0.0f) ? sums[k * DIM + d] / fmaxf(c, 1.0f) : old;
}

// ---------------------------------------------------------------------------
// One-hot output, (32,4096,100) f32.  grid: NPTS*100/256.
// ---------------------------------------------------------------------------
__global__ void km_onehot(const int* __restrict__ assign, float* __restrict__ out) {
    int idx = blockIdx.x * 256 + threadIdx.x;
    if (idx >= NPTS * NCLUST) return;
    int p = idx / NCLUST;
    int c = idx - p * NCLUST;
    out[idx] = (assign[p] == c) ? 1.0f : 0.0f;
}

// ---------------------------------------------------------------------------
extern "C" void kernel_launch(void* const* d_in, const int* in_sizes, int n_in,
                              void* d_out, int out_size, void* d_ws, size_t ws_size,
                              hipStream_t stream) {
    (void)in_sizes; (void)n_in; (void)out_size; (void)ws_size;
    const float* x = (const float*)d_in[0];
    float* ws      = (float*)d_ws;

    // workspace layout (floats)
    float* centers = ws;                 // 112*128 = 14336
    float* sums    = ws + 14336;         // 112*128 (100 used)
    float* c2      = ws + 28672;         // 112
    float* counts  = ws + 28784;         // 112
    int*   assign  = (int*)(ws + 28928); // 131072 ints
    float* out     = (float*)d_out;

    km_init_centers<<<KPAD, DIM, 0, stream>>>(x, centers);

    for (int it = 0; it < 10; ++it) {
        km_c2_zero<<<KPAD, DIM, 0, stream>>>(centers, c2, sums, counts);
        km_assign<<<NPTS / 64, 128, 0, stream>>>(x, centers, c2, assign);
        km_accum<<<NPTS / 512, 256, 0, stream>>>(x, assign, sums, counts);
        km_update<<<NCLUST, DIM, 0, stream>>>(sums, counts, centers);
    }

    km_c2_zero<<<KPAD, DIM, 0, stream>>>(centers, c2, sums, counts);
    km_assign<<<NPTS / 64, 128, 0, stream>>>(x, centers, c2, assign);
    km_onehot<<<(NPTS * NCLUST) / 256, 256, 0, stream>>>(assign, out);
}